// KANLayer_48902497632477
// MI455X (gfx1250) — compile-verified
//
#include <hip/hip_runtime.h>
#include <math.h>

typedef __attribute__((ext_vector_type(2))) float v2f;
typedef __attribute__((ext_vector_type(8))) float v8f;

#define BATCH   1024
#define N_IN    512
#define N_OUT   256
#define GK      257        // GRID_SIZE + K = 254 + 3
#define TB      16         // batch rows per block
#define IC      64         // K-chunk (input-dim chunk)
#define NCHUNK  (N_IN / IC)
#define LNS     260        // padded LDS stride for LN tile

__launch_bounds__(256)
__global__ void kan_fused_kernel(const float* __restrict__ x,
                                 const float* __restrict__ W,
                                 const float* __restrict__ SW,
                                 const float* __restrict__ prelu_a,
                                 float* __restrict__ out)
{
    __shared__ float  As[IC][TB];     // tanh(x) chunk, [k][m] layout for WMMA A fetch
    __shared__ float4 Tw[TB][IC];     // shifted 4-tap spline weights
    __shared__ int    Tj[TB][IC];     // clamped gather base index (0..253)
    __shared__ float  Ys[TB][LNS];    // y tile for LayerNorm
    __shared__ float  redS[TB][16];
    __shared__ float  redQ[TB][16];
    __shared__ float  rowMu[TB];
    __shared__ float  rowRs[TB];

    const int tid   = threadIdx.x;
    const int lane  = tid & 31;
    const int wv    = tid >> 5;        // wave id 0..7 -> output cols [32*wv, 32*wv+32)
    const int half  = lane >> 4;       // 0/1
    const int ln    = lane & 15;
    const int bbase = blockIdx.x * TB;

    const float h    = 2.0f / 254.0f;
    const float g0   = -1.0f - 3.0f * h;         // first knot
    const float invh = 254.0f / 2.0f;
    const float topx = g0 + 260.0f * h;          // last knot g[260]

    v8f c0 = {};   // accumulator: rows 8*half+r, col = 32*wv + ln
    v8f c1 = {};   // accumulator: rows 8*half+r, col = 32*wv + 16 + ln

    for (int ch = 0; ch < NCHUNK; ++ch) {
        const int i0 = ch * IC;
        __syncthreads();   // protect LDS reuse from previous chunk

        // ---- cooperative: tanh + uniform-knot cubic B-spline taps for chunk ----
        #pragma unroll
        for (int q = 0; q < (TB * IC) / 256; ++q) {
            int idx = tid + q * 256;             // 0..1023
            int br  = idx >> 6;                  // 0..15
            int ii  = idx & (IC - 1);            // 0..63
            float xv = x[(size_t)(bbase + br) * N_IN + (i0 + ii)];
            As[ii][br] = tanhf(xv);

            float4 v = make_float4(0.f, 0.f, 0.f, 0.f);
            int jbc = 0;
            if (xv >= g0 && xv < topx) {
                float s = (xv - g0) * invh;
                int   m = (int)floorf(s);
                m = min(max(m, 0), 259);
                float u  = s - (float)m;
                float um = 1.0f - u;
                float u2 = u * u, u3 = u2 * u;
                float wt[4];
                wt[0] = um * um * um * (1.0f / 6.0f);
                wt[1] = (3.0f * u3 - 6.0f * u2 + 4.0f) * (1.0f / 6.0f);
                wt[2] = (-3.0f * u3 + 3.0f * u2 + 3.0f * u + 1.0f) * (1.0f / 6.0f);
                wt[3] = u3 * (1.0f / 6.0f);
                int jb = m - 3;                  // first tap index, may be <0 or >253
                jbc    = min(max(jb, 0), GK - 4);
                int sh = jb - jbc;               // -3..3
                float vt[4];
                #pragma unroll
                for (int tp = 0; tp < 4; ++tp) {
                    int t = tp - sh;
                    vt[tp] = (t >= 0 && t <= 3) ? wt[t] : 0.0f;
                }
                v = make_float4(vt[0], vt[1], vt[2], vt[3]);
            }
            Tw[br][ii] = v;
            Tj[br][ii] = jbc;
        }
        __syncthreads();

        // ---- dense part: b_out += tanh(x)_tile @ W^T_tile via f32 WMMA ----
        const int o0 = 32 * wv + ln;
        const int o1 = o0 + 16;
        #pragma unroll 4
        for (int kk = 0; kk < IC / 4; ++kk) {
            int kl = kk * 4 + 2 * half;          // this lane's K pair
            int kg = i0 + kl;
            v2f a, b0, b1;
            a.x  = As[kl][ln];
            a.y  = As[kl + 1][ln];
            b0.x = W[(size_t)o0 * N_IN + kg];
            b0.y = W[(size_t)o0 * N_IN + kg + 1];
            b1.x = W[(size_t)o1 * N_IN + kg];
            b1.y = W[(size_t)o1 * N_IN + kg + 1];
            c0 = __builtin_amdgcn_wmma_f32_16x16x4_f32(false, a, false, b0,
                                                       (short)0, c0, false, false);
            c1 = __builtin_amdgcn_wmma_f32_16x16x4_f32(false, a, false, b1,
                                                       (short)0, c1, false, false);
        }

        // ---- sparse part: 4-tap spline gather (SW resident in 192MB L2) ----
        {
            const float* sw0 = SW + (size_t)o0 * N_IN * GK;
            const float* sw1 = SW + (size_t)o1 * N_IN * GK;
            for (int ii = 0; ii < IC; ++ii) {
                size_t ibase = (size_t)(i0 + ii) * GK;
                #pragma unroll
                for (int r = 0; r < 8; ++r) {
                    int br = 8 * half + r;
                    float4 w4 = Tw[br][ii];
                    if (w4.x + w4.y + w4.z + w4.w != 0.0f) {   // skip out-of-domain x
                        size_t off = ibase + (size_t)Tj[br][ii];
                        const float* p0 = sw0 + off;
                        const float* p1 = sw1 + off;
                        c0[r] += w4.x * p0[0] + w4.y * p0[1] + w4.z * p0[2] + w4.w * p0[3];
                        c1[r] += w4.x * p1[0] + w4.y * p1[1] + w4.z * p1[2] + w4.w * p1[3];
                    }
                }
            }
        }
    }

    // ---- write y tile to LDS for fused LayerNorm ----
    {
        int ob = 32 * wv + ln;
        #pragma unroll
        for (int r = 0; r < 8; ++r) {
            int br = 8 * half + r;
            Ys[br][ob]      = c0[r];
            Ys[br][ob + 16] = c1[r];
        }
    }
    __syncthreads();

    // ---- LayerNorm over 256 outputs per row + PReLU ----
    const int row = tid >> 4;    // 0..15
    const int col = tid & 15;    // 0..15
    float s = 0.f, q = 0.f;
    #pragma unroll
    for (int j = 0; j < 16; ++j) {
        float vy = Ys[row][col + 16 * j];
        s += vy;
        q += vy * vy;
    }
    redS[row][col] = s;
    redQ[row][col] = q;
    __syncthreads();
    if (col == 0) {
        float ts = 0.f, tq = 0.f;
        #pragma unroll
        for (int j = 0; j < 16; ++j) { ts += redS[row][j]; tq += redQ[row][j]; }
        float mu  = ts * (1.0f / 256.0f);
        float var = tq * (1.0f / 256.0f) - mu * mu;
        rowMu[row] = mu;
        rowRs[row] = rsqrtf(var + 1e-5f);
    }
    __syncthreads();
    const float aco = prelu_a[0];
    const float mu  = rowMu[row];
    const float rs  = rowRs[row];
    #pragma unroll
    for (int j = 0; j < 16; ++j) {
        int o = col + 16 * j;
        float vy = (Ys[row][o] - mu) * rs;
        out[(size_t)(bbase + row) * N_OUT + o] = (vy >= 0.f) ? vy : aco * vy;
    }
}

extern "C" void kernel_launch(void* const* d_in, const int* in_sizes, int n_in,
                              void* d_out, int out_size, void* d_ws, size_t ws_size,
                              hipStream_t stream)
{
    const float* x  = (const float*)d_in[0];   // (1024, 512)
    const float* W  = (const float*)d_in[1];   // (256, 512)
    const float* SW = (const float*)d_in[2];   // (256, 512, 257)
    const float* pa = (const float*)d_in[3];   // (1,)
    float* out = (float*)d_out;                // (1024, 256)

    kan_fused_kernel<<<dim3(BATCH / TB), dim3(256), 0, stream>>>(x, W, SW, pa, out);
}